// InpatientInput_41815801594422
// MI455X (gfx1250) — compile-verified
//
#include <hip/hip_runtime.h>

#define THREADS 256
#define MAX_BLOCKS 2048
#define SPLIT 32

// staging geometry: per buffer, per array: THREADS lanes * 16B (one B128 each)
#define CHUNK_BYTES (THREADS * 16)        // 4096 B per array per buffer
#define BUF_BYTES   (4 * CHUNK_BYTES)     // 16 KB per buffer (4 arrays)
#define STAGE_BYTES (2 * BUF_BYTES)       // 32 KB double-buffered

__global__ void zero_f32_kernel(float* __restrict__ p, int n) {
    int i = blockIdx.x * blockDim.x + threadIdx.x;
    if (i < n) p[i] = 0.0f;
}

// Issue one async global->LDS 128-bit copy (per-lane), tracked by ASYNCcnt.
__device__ __forceinline__ void async_copy_b128(unsigned lds_off, const void* gaddr) {
    asm volatile("global_load_async_to_lds_b128 %0, %1, off"
                 :: "v"(lds_off), "v"(gaddr)
                 : "memory");
}

__device__ __forceinline__ void wait_async_le4() {
    asm volatile("s_wait_asynccnt 0x4" ::: "memory");
}
__device__ __forceinline__ void wait_async_0() {
    asm volatile("s_wait_asynccnt 0x0" ::: "memory");
}

// Streaming histogram: async-stage event data into LDS (double buffered),
// consume from LDS into a block-private LDS histogram.
// use_atomic==0: write per-block partial histogram to dst + blockIdx.x*size.
// use_atomic==1: global-atomic-add the LDS histogram into dst (len size).
__global__ __launch_bounds__(THREADS)
void hist_kernel(const int*   __restrict__ idx,
                 const float* __restrict__ rate,
                 const float* __restrict__ st,
                 const float* __restrict__ en,
                 const float* __restrict__ tptr,
                 float*       __restrict__ dst,
                 int n, int size, int use_atomic) {
    extern __shared__ char smem[];
    float* hist = (float*)smem;
    // histogram occupies size floats; stage region follows (size*4 is 16B-aligned
    // for size multiple of 4; round up defensively)
    unsigned hist_bytes = ((unsigned)size * 4u + 15u) & ~15u;
    char* stage = smem + hist_bytes;
    const unsigned stage_off = (unsigned)(uintptr_t)stage;   // LDS byte offset

    for (int i = threadIdx.x; i < size; i += THREADS) hist[i] = 0.0f;
    __syncthreads();

    const float t = tptr[0];

    const int n4 = n >> 2;                       // number of full vec4 groups
    const char* gidx = (const char*)idx;
    const char* grat = (const char*)rate;
    const char* gst  = (const char*)st;
    const char* gen  = (const char*)en;

    const int stride = gridDim.x * THREADS;
    const int g0 = blockIdx.x * THREADS + threadIdx.x;
    const unsigned lane_off = (unsigned)threadIdx.x * 16u;

    if (n4 > 0) {
        // ---- prologue: stage buffer 0 for group g0 (clamped) ----
        {
            int gc = (g0 < n4) ? g0 : (n4 - 1);
            size_t gb = (size_t)gc * 16u;
            unsigned lb = stage_off + 0u * BUF_BYTES + lane_off;
            async_copy_b128(lb + 0u * CHUNK_BYTES, gidx + gb);
            async_copy_b128(lb + 1u * CHUNK_BYTES, grat + gb);
            async_copy_b128(lb + 2u * CHUNK_BYTES, gst  + gb);
            async_copy_b128(lb + 3u * CHUNK_BYTES, gen  + gb);
        }

        unsigned buf = 0;
        for (int g = g0; g < n4; g += stride) {
            // ---- stage next chunk into the other buffer ----
            {
                int gn = g + stride;
                int gc = (gn < n4) ? gn : (n4 - 1);
                size_t gb = (size_t)gc * 16u;
                unsigned lb = stage_off + (buf ^ 1u) * BUF_BYTES + lane_off;
                async_copy_b128(lb + 0u * CHUNK_BYTES, gidx + gb);
                async_copy_b128(lb + 1u * CHUNK_BYTES, grat + gb);
                async_copy_b128(lb + 2u * CHUNK_BYTES, gst  + gb);
                async_copy_b128(lb + 3u * CHUNK_BYTES, gen  + gb);
            }

            // oldest 4 async ops (current buffer) are complete when cnt<=4
            wait_async_le4();

            // ---- consume current buffer from LDS ----
            const char* sb = stage + buf * BUF_BYTES + threadIdx.x * 16u;
            int4   vi = *(const int4*)  (sb + 0u * CHUNK_BYTES);
            float4 vr = *(const float4*)(sb + 1u * CHUNK_BYTES);
            float4 vs = *(const float4*)(sb + 2u * CHUNK_BYTES);
            float4 ve = *(const float4*)(sb + 3u * CHUNK_BYTES);

            if (vs.x <= t && t < ve.x) atomicAdd(&hist[vi.x], vr.x);
            if (vs.y <= t && t < ve.y) atomicAdd(&hist[vi.y], vr.y);
            if (vs.z <= t && t < ve.z) atomicAdd(&hist[vi.z], vr.z);
            if (vs.w <= t && t < ve.w) atomicAdd(&hist[vi.w], vr.w);

            buf ^= 1u;
        }
        // quiesce outstanding async ops before LDS reuse / exit
        wait_async_0();
    }

    // scalar tail (n not divisible by 4)
    for (int i = (n4 << 2) + g0; i < n; i += stride) {
        float s = st[i], e = en[i];
        if (s <= t && t < e) atomicAdd(&hist[idx[i]], rate[i]);
    }

    __syncthreads();

    if (use_atomic) {
        for (int i = threadIdx.x; i < size; i += THREADS) {
            float v = hist[i];
            if (v != 0.0f) atomicAdd(&dst[i], v);
        }
    } else {
        float* mine = dst + (size_t)blockIdx.x * (size_t)size;
        for (int i = threadIdx.x; i < size; i += THREADS) mine[i] = hist[i];
    }
}

// Reduce nb partial histograms of length `size` into out (out pre-zeroed).
// SPLIT-way parallelism over the partial dimension; consecutive threads read
// consecutive bins -> fully coalesced.
__global__ void reduce_partials_kernel(const float* __restrict__ partials,
                                       float* __restrict__ out,
                                       int nb, int size) {
    int tid   = blockIdx.x * blockDim.x + threadIdx.x;
    int total = size * SPLIT;
    if (tid >= total) return;
    int bin = tid % size;
    int s   = tid / size;
    float acc = 0.0f;
    for (int j = s; j < nb; j += SPLIT)
        acc += partials[(size_t)j * (size_t)size + bin];
    atomicAdd(&out[bin], acc);
}

extern "C" void kernel_launch(void* const* d_in, const int* in_sizes, int n_in,
                              void* d_out, int out_size, void* d_ws, size_t ws_size,
                              hipStream_t stream) {
    const int*   idx  = (const int*)  d_in[0];
    const float* rate = (const float*)d_in[1];
    const float* st   = (const float*)d_in[2];
    const float* en   = (const float*)d_in[3];
    const float* tptr = (const float*)d_in[4];

    const int n    = in_sizes[0];
    const int size = out_size;          // == SIZE (2000)
    float* out = (float*)d_out;

    const size_t per_hist = (size_t)size * sizeof(float);
    int maxB = (per_hist > 0) ? (int)(ws_size / per_hist) : 0;
    int B = MAX_BLOCKS;
    if (B > maxB) B = maxB;

    const size_t hist_bytes = (per_hist + 15) & ~(size_t)15;
    const size_t shmem = hist_bytes + STAGE_BYTES;

    const int zero_blocks = (size + THREADS - 1) / THREADS;
    zero_f32_kernel<<<zero_blocks, THREADS, 0, stream>>>(out, size);

    if (B >= 64) {
        // partial-histogram path: no global atomic hot-spotting
        hist_kernel<<<B, THREADS, shmem, stream>>>(
            idx, rate, st, en, tptr, (float*)d_ws, n, size, /*use_atomic=*/0);
        const int total = size * SPLIT;
        const int rblocks = (total + THREADS - 1) / THREADS;
        reduce_partials_kernel<<<rblocks, THREADS, 0, stream>>>(
            (float*)d_ws, out, B, size);
    } else {
        // fallback: direct global atomics from each block's LDS histogram
        hist_kernel<<<1024, THREADS, shmem, stream>>>(
            idx, rate, st, en, tptr, out, n, size, /*use_atomic=*/1);
    }
}